// BiorUpSampling_56504589746636
// MI455X (gfx1250) — compile-verified
//
#include <hip/hip_runtime.h>

typedef float v2f __attribute__((ext_vector_type(2)));
typedef float v8f __attribute__((ext_vector_type(8)));

__device__ __constant__ float c_h[9] = {
    0.03782845550699535f, -0.02384946501937986f, -0.1106244044184226f,
    0.3774028556126536f,  0.8526986790094022f,   0.3774028556126537f,
    -0.1106244044184226f, -0.02384946501937986f, 0.03782845550699535f};

// Branchless tap: clamped unconditional load + select (no exec-mask branches).
__device__ __forceinline__ float tap(int t) {
    int tc = t < 0 ? 0 : (t > 8 ? 8 : t);
    float raw = c_h[tc];
    return (t >= 0 && t <= 8) ? raw : 0.0f;
}

// One wave32 computes one 16x16 output tile:
//   Out = A * In * A^T with A[y][i] = h[y+8-2i] (support: 12 input rows per
//   16 output rows), via two chains of 3x V_WMMA_F32_16X16X4_F32 (K=12),
// with a per-wave LDS transpose between the H-pass and the W-pass.
__global__ __launch_bounds__(256) void bior_up_wmma(const float* __restrict__ in,
                                                    float* __restrict__ out) {
    const int lane = threadIdx.x & 31;
    const int wave = threadIdx.x >> 5;
    const int tile = blockIdx.x * 8 + wave;     // 65536 tiles total
    const int b  = tile >> 12;                  // 4096 tiles per image (64x64)
    const int tr = (tile >> 6) & 63;
    const int tc = tile & 63;

    const int m  = lane & 15;   // M index (A rows) or N index (B/C cols)
    const int kh = lane >> 4;   // which K-half of a K=4 chunk this lane holds

    // Filter chunks: A-operand (16x4) and B-operand (4x16) layouts coincide
    // (lane = kh*16 + idx, vgpr = k'%2), so w[] serves both stages.
    // Only K=0..11 is nonzero -> 3 chunks.
    v2f w[3];
    #pragma unroll
    for (int kk = 0; kk < 3; ++kk) {
        const int i0 = 4 * kk + 2 * kh;
        w[kk].x = tap(m + 8 - 2 * (i0 + 0));
        w[kk].y = tap(m + 8 - 2 * (i0 + 1));
    }

    // Load the 12x16 input block (rows R/2-2 .. R/2+9) in B layout.
    // Branchless: clamp address, unconditional load, select 0 for OOB.
    const int r0 = tr * 8 - 2;
    const int c0 = tc * 8 - 2;
    const float* inb = in + (size_t)b * (512 * 512);
    const int gc  = c0 + m;
    const int ccl = gc < 0 ? 0 : (gc > 511 ? 511 : gc);
    const bool cok = (gc >= 0) && (gc < 512);
    v2f bi[3];
    #pragma unroll
    for (int kk = 0; kk < 3; ++kk) {
        float vv[2];
        #pragma unroll
        for (int v = 0; v < 2; ++v) {
            const int gr  = r0 + 4 * kk + 2 * kh + v;
            const int rcl = gr < 0 ? 0 : (gr > 511 ? 511 : gr);
            const float raw = inb[rcl * 512 + ccl];
            vv[v] = (cok && gr >= 0 && gr < 512) ? raw : 0.0f;
        }
        bi[kk].x = vv[0];
        bi[kk].y = vv[1];
    }

    // Stage 1 (H pass): T = A * In, accumulated over three K=4 chunks.
    v8f t = {};
    #pragma unroll
    for (int kk = 0; kk < 3; ++kk)
        t = __builtin_amdgcn_wmma_f32_16x16x4_f32(false, w[kk], false, bi[kk],
                                                  (short)0, t, false, false);

    // Transpose T (C/D layout -> A layout) through this wave's LDS slab.
    // Same-wave LDS ops are in-order; fence compiler + DScnt only.
    __shared__ float lds[8][16 * 17];
    #pragma unroll
    for (int v = 0; v < 8; ++v)
        lds[wave][(kh * 8 + v) * 17 + m] = t[v];
    __builtin_amdgcn_wave_barrier();
    asm volatile("s_wait_dscnt 0" ::: "memory");

    v2f at[3];
    #pragma unroll
    for (int kk = 0; kk < 3; ++kk) {
        const int j0 = 4 * kk + 2 * kh;
        at[kk].x = lds[wave][m * 17 + j0 + 0];
        at[kk].y = lds[wave][m * 17 + j0 + 1];
    }

    // Stage 2 (W pass): Out = T * A^T (K=12 again).
    v8f o = {};
    #pragma unroll
    for (int kk = 0; kk < 3; ++kk)
        o = __builtin_amdgcn_wmma_f32_16x16x4_f32(false, at[kk], false, w[kk],
                                                  (short)0, o, false, false);

    // Store the 16x16 output tile (C/D layout: row = kh*8+v, col = m).
    float* ob = out + (size_t)b * (1024 * 1024)
                    + (size_t)(tr * 16) * 1024 + (tc * 16);
    #pragma unroll
    for (int v = 0; v < 8; ++v)
        ob[(kh * 8 + v) * 1024 + m] = o[v];
}

extern "C" void kernel_launch(void* const* d_in, const int* in_sizes, int n_in,
                              void* d_out, int out_size, void* d_ws, size_t ws_size,
                              hipStream_t stream) {
    const float* in = (const float*)d_in[0];
    float* out = (float*)d_out;
    // 16 images * 64*64 tiles = 65536 tiles; 8 waves (tiles) per 256-thread block.
    dim3 grid(8192), block(256);
    hipLaunchKernelGGL(bior_up_wmma, grid, block, 0, stream, in, out);
}